// TopKAccuracy_27084063768743
// MI455X (gfx1250) — compile-verified
//
#include <hip/hip_runtime.h>
#include <stdint.h>

// ---------------------------------------------------------------------------
// Top-K(30) overlap + sorted-value L1 diff over [131072 x 1024] fp32 rows.
// One wave32 per row; async global->LDS double-buffered row streaming (CDNA5
// GLOBAL_LOAD_ASYNC_TO_LDS_B128 + s_wait_asynccnt); threshold + branchless
// ballot compaction + cross-lane bitonic selection. Deterministic 2-stage
// reduction (no float atomics).
// ---------------------------------------------------------------------------

#define KTOP            30
#define COLS            1024
#define WAVES_PER_BLOCK 4
#define BLOCKS          2048
#define CAND_CAP        256
#define NWAVES          (BLOCKS * WAVES_PER_BLOCK)

__device__ __forceinline__ unsigned lane_id() { return threadIdx.x & 31u; }

// ---- cross-lane bitonic helpers (wave32) ----------------------------------

__device__ __forceinline__ void cmpswap_kv(float& v, int& ix, int j, bool keep_max) {
  float ov = __shfl_xor(v, j, 32);
  int   oi = __shfl_xor(ix, j, 32);
  bool take = keep_max ? (ov > v) : (ov < v);
  v  = take ? ov : v;
  ix = take ? oi : ix;
}

__device__ __forceinline__ void bitonic_sort_desc_kv(float& v, int& ix, unsigned lane) {
#pragma unroll
  for (int k = 2; k <= 32; k <<= 1) {
#pragma unroll
    for (int j = k >> 1; j > 0; j >>= 1) {
      bool up = ((lane & (unsigned)k) == 0u);
      bool keep_max = (up == ((lane & (unsigned)j) == 0u));
      cmpswap_kv(v, ix, j, keep_max);
    }
  }
}

__device__ __forceinline__ void bitonic_merge_desc_kv(float& v, int& ix, unsigned lane) {
#pragma unroll
  for (int j = 16; j > 0; j >>= 1)
    cmpswap_kv(v, ix, j, (lane & (unsigned)j) == 0u);
}

__device__ __forceinline__ void bitonic_sort_desc_v(float& v, unsigned lane) {
#pragma unroll
  for (int k = 2; k <= 32; k <<= 1) {
#pragma unroll
    for (int j = k >> 1; j > 0; j >>= 1) {
      bool up = ((lane & (unsigned)k) == 0u);
      bool keep_max = (up == ((lane & (unsigned)j) == 0u));
      float ov = __shfl_xor(v, j, 32);
      v = keep_max ? fmaxf(v, ov) : fminf(v, ov);
    }
  }
}

// ---- CDNA5 async global->LDS row copy (8 x b128 = 4KB per wave) -----------

__device__ __forceinline__ void async_row_to_lds(unsigned lds_base, const float* gptr,
                                                 unsigned lane) {
  unsigned l  = lds_base + lane * 16u;
  uint64_t ga = (uint64_t)(uintptr_t)gptr + (uint64_t)(lane * 16u);
  // INST_OFFSET is added to both the LDS and the global address (ISA 10.4.4),
  // so one base pair covers the whole 4KB row.
  asm volatile(
      "global_load_async_to_lds_b128 %0, %1, off\n\t"
      "global_load_async_to_lds_b128 %0, %1, off offset:512\n\t"
      "global_load_async_to_lds_b128 %0, %1, off offset:1024\n\t"
      "global_load_async_to_lds_b128 %0, %1, off offset:1536\n\t"
      "global_load_async_to_lds_b128 %0, %1, off offset:2048\n\t"
      "global_load_async_to_lds_b128 %0, %1, off offset:2560\n\t"
      "global_load_async_to_lds_b128 %0, %1, off offset:3072\n\t"
      "global_load_async_to_lds_b128 %0, %1, off offset:3584"
      :: "v"(l), "v"(ga) : "memory");
}

__device__ __forceinline__ void wait_async_le8() {
  asm volatile("s_wait_asynccnt 8" ::: "memory");   // oldest 8 (in-order) done
}
__device__ __forceinline__ void wait_async_0() {
  asm volatile("s_wait_asynccnt 0" ::: "memory");
}

// ---- per-row top-30 (sorted desc): lane j<30 ends with j-th (val, idx) ----
// cand[] must have CAND_CAP + 32 entries (last 32 = per-lane spill slots).

__device__ __forceinline__ void topk_row(const float* buf, uint2* cand, unsigned lane,
                                         float& out_v, int& out_i) {
  // 32 elements per lane; element index of x[a] is (a/4)*128 + lane*4 + (a&3).
  float x[32];
  const float4* q4 = (const float4*)buf;
#pragma unroll
  for (int i = 0; i < 8; ++i) {
    float4 q = q4[i * 32 + lane];
    x[4 * i + 0] = q.x; x[4 * i + 1] = q.y; x[4 * i + 2] = q.z; x[4 * i + 3] = q.w;
  }

  // Lane-local max via 4 independent chains (depth ~9 instead of 31).
  float a0 = x[0], a1 = x[1], a2 = x[2], a3 = x[3];
#pragma unroll
  for (int a = 4; a < 32; a += 4) {
    a0 = fmaxf(a0, x[a + 0]);
    a1 = fmaxf(a1, x[a + 1]);
    a2 = fmaxf(a2, x[a + 2]);
    a3 = fmaxf(a3, x[a + 3]);
  }
  float lm = fmaxf(fmaxf(a0, a1), fmaxf(a2, a3));

  // Threshold T = 30th-largest lane max  =>  #elements >= T is provably >= 30
  // (the top-30 lane maxima are 30 distinct elements all >= T).
  float sm = lm;
  bitonic_sort_desc_v(sm, lane);
  float T = __shfl(sm, KTOP - 1, 32);

  // Branchless ballot compaction: every lane stores every element; misses and
  // overflow go to a per-lane spill slot at CAND_CAP + lane.
  unsigned cnt = 0;
  const unsigned spill = CAND_CAP + lane;
  const unsigned lane4 = lane * 4u;
#pragma unroll
  for (int a = 0; a < 32; ++a) {
    bool p = (x[a] >= T);
    unsigned m = (unsigned)__ballot(p);
    unsigned pos = __builtin_amdgcn_mbcnt_lo(m, cnt);  // cnt + popc(m & lt_mask)
    unsigned slot = (p && pos < CAND_CAP) ? pos : spill;
    unsigned ei = (unsigned)((a >> 2) * 128 + (a & 3)) + lane4;
    cand[slot] = make_uint2(__float_as_uint(x[a]), ei);
    cnt += (unsigned)__popc(m);
  }
  if (cnt > CAND_CAP) cnt = CAND_CAP;   // E[cnt]~74 for Gaussian rows

  // Chunked selection: running sorted top-32 across lanes.
  const float NEG = -3.402823466e38f;
  float v; int ix;
  if (lane < cnt) { uint2 e = cand[lane]; v = __uint_as_float(e.x); ix = (int)e.y; }
  else            { v = NEG; ix = 0x7fffffff; }
  bitonic_sort_desc_kv(v, ix, lane);

  for (unsigned base = 32; base < cnt; base += 32) {
    float cv; int ci;
    unsigned p2 = base + lane;
    if (p2 < cnt) { uint2 e = cand[p2]; cv = __uint_as_float(e.x); ci = (int)e.y; }
    else          { cv = NEG; ci = 0x7fffffff; }
    bitonic_sort_desc_kv(cv, ci, lane);
    // top-32 of two desc-sorted lists: pair lane j with lane 31-j, keep max
    // (bitonic half-cleaner), then 5-stage cleanup.
    float bv = __shfl_xor(cv, 31, 32);
    int   bi = __shfl_xor(ci, 31, 32);
    if (bv > v) { v = bv; ix = bi; }
    bitonic_merge_desc_kv(v, ix, lane);
  }
  out_v = v; out_i = ix;
}

// ---- main kernel: one wave per row, async double-buffered streaming -------

__global__ void __launch_bounds__(WAVES_PER_BLOCK * 32)
topk_main(const float* __restrict__ pred, const float* __restrict__ target,
          float* __restrict__ ws, int nrows) {
  __shared__ float s_buf[WAVES_PER_BLOCK][2][COLS];            // 32 KB
  __shared__ uint2 s_cand[WAVES_PER_BLOCK][CAND_CAP + 32];     //  9 KB

  const unsigned lane = lane_id();
  const unsigned wib  = threadIdx.x >> 5;
  const unsigned gw   = blockIdx.x * WAVES_PER_BLOCK + wib;
  const unsigned NW   = gridDim.x * WAVES_PER_BLOCK;

  float* buf0 = &s_buf[wib][0][0];
  float* buf1 = &s_buf[wib][1][0];
  uint2* cand = &s_cand[wib][0];
  // Generic (flat) shared pointers carry the raw LDS byte offset in addr[31:0].
  unsigned lds0 = (unsigned)(uintptr_t)buf0;
  unsigned lds1 = (unsigned)(uintptr_t)buf1;

  float diff_acc = 0.0f, match_acc = 0.0f;

  if ((int)gw < nrows) {
    async_row_to_lds(lds0, target + (size_t)gw * COLS, lane);
    for (int r = (int)gw; r < nrows; r += (int)NW) {
      async_row_to_lds(lds1, pred + (size_t)r * COLS, lane);   // overlap w/ tgt compute
      wait_async_le8();                                        // target[r] landed
      float tv; int ti;
      topk_row(buf0, cand, lane, tv, ti);

      int rn = (r + (int)NW < nrows) ? (r + (int)NW) : r;      // clamp stray prefetch
      async_row_to_lds(lds0, target + (size_t)rn * COLS, lane);
      wait_async_le8();                                        // pred[r] landed
      float pv; int pi;
      topk_row(buf1, cand, lane, pv, pi);

      // Overlap count: does pred's j-th index appear in target's top-30?
      bool found = false;
#pragma unroll
      for (int m = 0; m < KTOP; ++m) {
        int t = __shfl(ti, m, 32);        // uniform src -> readlane broadcast
        found = found || (pi == t);
      }
      if (lane < KTOP) {
        diff_acc  += fabsf(pv - tv);
        match_acc += found ? 1.0f : 0.0f;
      }
    }
    wait_async_0();
  }

  // Per-wave reduction, then one deterministic partial per wave.
#pragma unroll
  for (int j = 16; j > 0; j >>= 1) {
    diff_acc  += __shfl_xor(diff_acc, j, 32);
    match_acc += __shfl_xor(match_acc, j, 32);
  }
  if (lane == 0) {
    ws[gw]      = match_acc;
    ws[NW + gw] = diff_acc;
  }
}

// ---- final reduction: 8192 wave partials -> (acc, diff) -------------------

__global__ void __launch_bounds__(256)
topk_reduce(const float* __restrict__ ws, float* __restrict__ out, int nw, int nrows) {
  __shared__ float sm[256], sd[256];
  float a = 0.0f, d = 0.0f;
  for (int i = threadIdx.x; i < nw; i += 256) { a += ws[i]; d += ws[nw + i]; }
  sm[threadIdx.x] = a; sd[threadIdx.x] = d;
  __syncthreads();
  for (int s = 128; s > 0; s >>= 1) {
    if ((int)threadIdx.x < s) {
      sm[threadIdx.x] += sm[threadIdx.x + s];
      sd[threadIdx.x] += sd[threadIdx.x + s];
    }
    __syncthreads();
  }
  if (threadIdx.x == 0) {
    out[0] = sm[0] / ((float)KTOP * (float)nrows);  // acc = mean(count/K)
    out[1] = sd[0] / (float)KTOP;                   // diff = sum|.| / K
  }
}

// ---------------------------------------------------------------------------

extern "C" void kernel_launch(void* const* d_in, const int* in_sizes, int n_in,
                              void* d_out, int out_size, void* d_ws, size_t ws_size,
                              hipStream_t stream) {
  (void)n_in; (void)out_size; (void)ws_size;
  const float* pred   = (const float*)d_in[0];
  const float* target = (const float*)d_in[1];
  float* out = (float*)d_out;
  float* ws  = (float*)d_ws;                 // needs 2*NWAVES*4 = 64 KB
  int nrows = in_sizes[0] / COLS;

  topk_main<<<BLOCKS, WAVES_PER_BLOCK * 32, 0, stream>>>(pred, target, ws, nrows);
  topk_reduce<<<1, 256, 0, stream>>>(ws, out, NWAVES, nrows);
}